// LSTM_31379031064854
// MI455X (gfx1250) — compile-verified
//
#include <hip/hip_runtime.h>

// LSTM fused kernel for MI455X (gfx1250, wave32, WMMA).
// B=512, T=512, I=64, H=50 (4H=200), O=10.
// One workgroup of 13 waves (416 threads) owns 16 batch rows and runs the
// full T=512 recurrence locally. Gates GEMM via V_WMMA_F32_16X16X4_F32.
// Two independent WMMA accumulator chains (x-part / h-part) to halve the
// dependent-WMMA critical path; summed at the end of each step.

#define B_   512
#define T_   512
#define I_   64
#define H_   50
#define G_   200   // 4*H
#define GP_  208   // gates N padded to 13*16
#define HP_  52    // H padded to multiple of 4 (WMMA K)
#define O_   10
#define KS_X 16    // I/4  k-steps for x @ W_ih^T
#define KS_H 13    // HP/4 k-steps for h @ W_hh^T
#define NT_  416   // 13 waves * 32

typedef float v2f __attribute__((ext_vector_type(2)));
typedef float v8f __attribute__((ext_vector_type(8)));

__device__ __forceinline__ v8f wmma_f32x4(v2f a, v2f b, v8f c) {
  // D = A(16x4) * B(4x16) + C(16x16), fp32
  return __builtin_amdgcn_wmma_f32_16x16x4_f32(false, a, false, b,
                                               (short)0, c, false, false);
}

__device__ __forceinline__ float frcp(float x) { return __builtin_amdgcn_rcpf(x); }
__device__ __forceinline__ float sigf(float x) { return frcp(1.0f + __expf(-x)); }
__device__ __forceinline__ float tanh_fast(float x) {
  return 1.0f - 2.0f * frcp(__expf(2.0f * x) + 1.0f);
}

__global__ __launch_bounds__(NT_) void lstm_fused_kernel(
    const float* __restrict__ x,     // [B, T, I]
    const float* __restrict__ Wih,   // [4H, I]
    const float* __restrict__ Whh,   // [4H, H]
    const float* __restrict__ bih,   // [4H]
    const float* __restrict__ bhh,   // [4H]
    const float* __restrict__ Wlin,  // [O, H]
    const float* __restrict__ blin,  // [O]
    float* __restrict__ out)         // [B, T, O]
{
  __shared__ float sh_h[16 * HP_];      // h_t, K-padded A-matrix source
  __shared__ float sh_gates[16 * GP_];  // WMMA results (pre-bias gates)
  __shared__ float sh_bias[G_];         // b_ih + b_hh
  __shared__ float sh_wl[O_ * H_];
  __shared__ float sh_bl[O_];

  const int tid  = threadIdx.x;
  const int lane = tid & 31;
  const int wave = tid >> 5;            // 0..12 -> N-tile
  const int n0   = wave << 4;
  const int mrow = lane & 15;           // M (A) / N (B) index within tile
  const int kh   = (lane >> 4) << 1;    // K half-select: 0 or 2
  const int b0   = blockIdx.x << 4;     // 16 batch rows per workgroup
  const int ncol = n0 + mrow;           // global gate column for B frags / stores
  const v2f vzero = {0.0f, 0.0f};

  // ---- one-time init ----
  for (int i = tid; i < 16 * HP_; i += NT_) sh_h[i] = 0.0f;     // h0 = 0 (+K pad)
  for (int i = tid; i < G_;       i += NT_) sh_bias[i] = bih[i] + bhh[i];
  for (int i = tid; i < O_ * H_;  i += NT_) sh_wl[i] = Wlin[i];
  if (tid < O_) sh_bl[tid] = blin[tid];

  // B fragments of W_ih^T (B[k][n] = Wih[n][k]) -- resident in VGPRs
  v2f bx[KS_X];
#pragma unroll
  for (int ks = 0; ks < KS_X; ++ks) {
    const int k = 4 * ks + kh;
    bx[ks] = (ncol < G_) ? *(const v2f*)(Wih + ncol * I_ + k) : vzero;
  }
  // B fragments of W_hh^T (B[k][n] = Whh[n][k]), K padded 50->52 with zeros
  v2f bh[KS_H];
#pragma unroll
  for (int ks = 0; ks < KS_H; ++ks) {
    const int k = 4 * ks + kh;
    v2f v = vzero;
    if (ncol < G_) {
      if (k + 1 < H_)      v = *(const v2f*)(Whh + ncol * H_ + k);
      else if (k < H_)     { v2f t = vzero; t.x = Whh[ncol * H_ + k]; v = t; }
    }
    bh[ks] = v;
  }

  // A fragments of x_t (A[m][k] = x[b0+m][t][k]); prefetch t=0
  const float* xrow = x + ((size_t)(b0 + mrow) * T_) * I_;
  v2f ax[KS_X], axn[KS_X];
#pragma unroll
  for (int ks = 0; ks < KS_X; ++ks)
    ax[ks] = *(const v2f*)(xrow + 4 * ks + kh);

  // per-thread cell state: each thread owns <=2 of the 16x50 (m,j) elements
  const int e1 = tid;                 // 0..415
  const int m1 = e1 / H_, j1 = e1 - m1 * H_;
  const int e2 = tid + NT_;           // 416..831
  const int m2 = e2 / H_, j2 = e2 - m2 * H_;
  const bool has2 = (e2 < 16 * H_);
  float c_a = 0.0f, c_b = 0.0f;

  __syncthreads();

  for (int t = 0; t < T_; ++t) {
    // -- prefetch next timestep's x A-fragments (hidden behind WMMA+act) --
    const int tn = (t + 1 < T_) ? (t + 1) : t;
#pragma unroll
    for (int ks = 0; ks < KS_X; ++ks)
      axn[ks] = *(const v2f*)(xrow + tn * I_ + 4 * ks + kh);

    // -- gates = x_t @ Wih^T + h_t @ Whh^T --
    // Two independent accumulator chains so the hardware can interleave
    // WMMAs instead of stalling on a single 29-deep D->C dependency.
    v8f accX = {0.f, 0.f, 0.f, 0.f, 0.f, 0.f, 0.f, 0.f};
    v8f accH = {0.f, 0.f, 0.f, 0.f, 0.f, 0.f, 0.f, 0.f};
#pragma unroll
    for (int ks = 0; ks < KS_H; ++ks) {
      v2f ah = *(const v2f*)(sh_h + mrow * HP_ + 4 * ks + kh);  // ds_load_b64
      accX = wmma_f32x4(ax[ks], bx[ks], accX);
      accH = wmma_f32x4(ah, bh[ks], accH);
    }
#pragma unroll
    for (int ks = KS_H; ks < KS_X; ++ks)
      accX = wmma_f32x4(ax[ks], bx[ks], accX);

    // C/D layout: element (M = v + 8*(lane>>4), N = ncol)
#pragma unroll
    for (int v = 0; v < 8; ++v)
      sh_gates[(v + ((lane >> 4) << 3)) * GP_ + ncol] = accX[v] + accH[v];

    __syncthreads();  // gates published

    // -- activations: i,f,g,o -> c,h (PyTorch gate order) --
    {
      const float* gr = sh_gates + m1 * GP_ + j1;
      float gi = gr[0]   + sh_bias[j1];
      float gf = gr[50]  + sh_bias[j1 + 50];
      float gg = gr[100] + sh_bias[j1 + 100];
      float go = gr[150] + sh_bias[j1 + 150];
      float ii = sigf(gi), ff = sigf(gf), oo = sigf(go);
      c_a = ff * c_a + ii * tanh_fast(gg);
      sh_h[m1 * HP_ + j1] = oo * tanh_fast(c_a);
    }
    if (has2) {
      const float* gr = sh_gates + m2 * GP_ + j2;
      float gi = gr[0]   + sh_bias[j2];
      float gf = gr[50]  + sh_bias[j2 + 50];
      float gg = gr[100] + sh_bias[j2 + 100];
      float go = gr[150] + sh_bias[j2 + 150];
      float ii = sigf(gi), ff = sigf(gf), oo = sigf(go);
      c_b = ff * c_b + ii * tanh_fast(gg);
      sh_h[m2 * HP_ + j2] = oo * tanh_fast(c_b);
    }

    // rotate prefetched A fragments (waits on loadcnt here, fully hidden)
#pragma unroll
    for (int ks = 0; ks < KS_X; ++ks) ax[ks] = axn[ks];

    __syncthreads();  // h_t published

    // -- fused output projection: out[b, t, :] = h @ Wlin^T + blin --
    if (tid < 16 * O_) {
      const int m = tid / O_, oc = tid - m * O_;
      float a = sh_bl[oc];
#pragma unroll 10
      for (int j = 0; j < H_; ++j)
        a += sh_h[m * HP_ + j] * sh_wl[oc * H_ + j];
      out[((size_t)(b0 + m) * T_ + t) * O_ + oc] = a;
    }
  }
}

extern "C" void kernel_launch(void* const* d_in, const int* in_sizes, int n_in,
                              void* d_out, int out_size, void* d_ws, size_t ws_size,
                              hipStream_t stream) {
  (void)in_sizes; (void)n_in; (void)out_size; (void)d_ws; (void)ws_size;
  const float* x    = (const float*)d_in[0];
  const float* Wih  = (const float*)d_in[1];
  const float* Whh  = (const float*)d_in[2];
  const float* bih  = (const float*)d_in[3];
  const float* bhh  = (const float*)d_in[4];
  const float* Wlin = (const float*)d_in[5];
  const float* blin = (const float*)d_in[6];
  float* out = (float*)d_out;

  lstm_fused_kernel<<<dim3(B_ / 16), dim3(NT_), 0, stream>>>(
      x, Wih, Whh, bih, bhh, Wlin, blin, out);
}